// HGT_64802466562492
// MI455X (gfx1250) — compile-verified
//
#include <hip/hip_runtime.h>

#define HIDD 32
#define HEADD 32
#define INF_ 128

typedef __attribute__((ext_vector_type(2))) float v2f;
typedef __attribute__((ext_vector_type(8))) float v8f;

// ---------------- WMMA f32 16x16 tile helper (V_WMMA_F32_16X16X4_F32) -------
// A: row-major [.., lda], rows row0..row0+15, K contiguous
// B: row-major [K, ldb], cols col0..col0+15
// A layout (ISA 7.12.2, 32-bit A 16x4): lane m=lane&15; v0 = A[m][k+2*hi], v1 = A[m][k+2*hi+1]
// B layout (row-striped): v0 = B[k+2*hi][col], v1 = B[k+2*hi+1][col]
__device__ __forceinline__ v8f wmma_tile_f32(const float* __restrict__ A, int lda, int row0,
                                             const float* __restrict__ B, int ldb, int col0,
                                             int K, int lane)
{
    int m  = lane & 15;
    int hi = lane >> 4;
    v8f c = {0.f, 0.f, 0.f, 0.f, 0.f, 0.f, 0.f, 0.f};
    const float* arow = A + (size_t)(row0 + m) * lda + 2 * hi;
    const float* bcol = B + (size_t)(2 * hi) * ldb + col0 + m;
    for (int k = 0; k < K; k += 4) {
        v2f a, b;
        a.x = arow[k];
        a.y = arow[k + 1];
        b.x = bcol[(size_t)k * ldb];
        b.y = bcol[(size_t)(k + 1) * ldb];
        c = __builtin_amdgcn_wmma_f32_16x16x4_f32(false, a, false, b, (short)0, c, false, false);
    }
    return c;
}

// monotonic float <-> uint encoding for atomicMax on floats
__device__ __forceinline__ unsigned fenc(float f) {
    unsigned u = __float_as_uint(f);
    return (u & 0x80000000u) ? ~u : (u | 0x80000000u);
}
__device__ __forceinline__ float fdec(unsigned u) {
    unsigned b = (u & 0x80000000u) ? (u & 0x7FFFFFFFu) : ~u;
    return __uint_as_float(b);
}

// ---------- combined relation weights: CW[kv*T+t] = W @ R[t]  (32x32 each) ---
__global__ void k_combine(const float* __restrict__ Wk, const float* __restrict__ Wv,
                          const float* __restrict__ Ratt, const float* __restrict__ Rmsg,
                          float* __restrict__ CW, int T)
{
    int mat = blockIdx.x;              // 0..2T-1
    int kv  = mat / T, t = mat % T;
    int c   = threadIdx.x >> 5;        // 0..31 (input dim)
    int e   = threadIdx.x & 31;        // 0..31 (output dim)
    const float* W = kv ? Wv : Wk;                                  // [HID, HEAD_D]
    const float* R = (kv ? Rmsg : Ratt) + (size_t)t * HIDD * HEADD; // [d, e]
    float acc = 0.f;
#pragma unroll
    for (int d = 0; d < HIDD; ++d)
        acc += W[c * HEADD + d] * R[d * HEADD + e];
    CW[(size_t)mat * HIDD * HEADD + c * HEADD + e] = acc;
}

// ---------------- zero init for amax / denom / agg ---------------------------
__global__ void k_init(unsigned* __restrict__ amax, float* __restrict__ denom,
                       float* __restrict__ agg, int N)
{
    int i = blockIdx.x * blockDim.x + threadIdx.x;
    if (i < N) { amax[i] = 0u; denom[i] = 0.f; }
    if (i < N * HIDD) agg[i] = 0.f;
}

// ---------------- input projection: h = x @ inW + b  (K=128) -----------------
__global__ void k_inproj(const float* __restrict__ x, const float* __restrict__ W,
                         const float* __restrict__ b, float* __restrict__ h, int N)
{
    int lane = threadIdx.x & 31;
    int tile = blockIdx.x * (blockDim.x >> 5) + (threadIdx.x >> 5);
    int row0 = tile * 16;
    if (row0 >= N) return;
    v8f c0 = wmma_tile_f32(x, INF_, row0, W, HIDD, 0, INF_, lane);
    v8f c1 = wmma_tile_f32(x, INF_, row0, W, HIDD, 16, INF_, lane);
    int n = lane & 15, hi = lane >> 4;
    float b0 = b[n], b1 = b[16 + n];
#pragma unroll
    for (int j = 0; j < 8; ++j) {
        int r = row0 + j + 8 * hi;
        h[(size_t)r * HIDD + n]      = c0[j] + b0;
        h[(size_t)r * HIDD + 16 + n] = c1[j] + b1;
    }
}

// ------- fused 7-way projection: q = h@Wq ; krel[t] = h@Ck[t] ; vrel[t]=h@Cv[t]
__global__ void k_proj7(const float* __restrict__ h, const float* __restrict__ Wq,
                        const float* __restrict__ CW,
                        float* __restrict__ q, float* __restrict__ krel,
                        float* __restrict__ vrel, int N, int T)
{
    int lane = threadIdx.x & 31;
    int tile = blockIdx.x * (blockDim.x >> 5) + (threadIdx.x >> 5);
    int row0 = tile * 16;
    if (row0 >= N) return;
    int m = blockIdx.y;                 // 0: q, 1..T: krel, T+1..2T: vrel
    const float* W;
    float* out;
    if (m == 0)      { W = Wq;                                    out = q; }
    else if (m <= T) { W = CW + (size_t)(m - 1) * HIDD * HEADD;   out = krel + (size_t)(m - 1) * N * HEADD; }
    else             { W = CW + (size_t)(m - 1) * HIDD * HEADD;   out = vrel + (size_t)(m - 1 - T) * N * HEADD; }
    v8f c0 = wmma_tile_f32(h, HIDD, row0, W, HEADD, 0, HIDD, lane);
    v8f c1 = wmma_tile_f32(h, HIDD, row0, W, HEADD, 16, HIDD, lane);
    int n = lane & 15, hi = lane >> 4;
#pragma unroll
    for (int j = 0; j < 8; ++j) {
        int r = row0 + j + 8 * hi;
        out[(size_t)r * HEADD + n]      = c0[j];
        out[(size_t)r * HEADD + 16 + n] = c1[j];
    }
}

// -------- wave-per-edge attention logit + segment max (atomic, enc-uint) -----
__global__ void k_edge_logit(const int* __restrict__ src, const int* __restrict__ dst,
                             const int* __restrict__ ety,
                             const float* __restrict__ krel, const float* __restrict__ q,
                             const float* __restrict__ pri_l,
                             float* __restrict__ logit, unsigned* __restrict__ amax,
                             int E, int N)
{
    int e = (blockIdx.x * blockDim.x + threadIdx.x) >> 5;
    if (e >= E) return;
    int lane = threadIdx.x & 31;
    int t = ety[e], s = src[e], d = dst[e];
    float kv = krel[((size_t)t * N + s) * HEADD + lane];
    float qv = q[(size_t)d * HEADD + lane];
    float v = kv * qv;
#pragma unroll
    for (int off = 16; off > 0; off >>= 1) v += __shfl_xor(v, off, 32);
    if (lane == 0) {
        float a = v * pri_l[t] * 0.17677669529663687f;   // 1/sqrt(32)
        logit[e] = a;
        atomicMax(&amax[d], fenc(a));
    }
}

// -------- thread-per-edge: ea = exp(a - amax[dst]); denom[dst] += ea ---------
__global__ void k_edge_exp(float* __restrict__ logit, const int* __restrict__ dst,
                           const unsigned* __restrict__ amax, float* __restrict__ denom,
                           int E)
{
    int e = blockIdx.x * blockDim.x + threadIdx.x;
    if (e >= E) return;
    int d = dst[e];
    float ea = __expf(logit[e] - fdec(amax[d]));
    logit[e] = ea;
    atomicAdd(&denom[d], ea);
}

// -------- wave-per-edge: agg[dst] += (ea/denom[dst]) * vrel[t][src] ----------
__global__ void k_edge_agg(const int* __restrict__ src, const int* __restrict__ dst,
                           const int* __restrict__ ety,
                           const float* __restrict__ vrel, const float* __restrict__ logit,
                           const float* __restrict__ denom, float* __restrict__ agg,
                           int E, int N)
{
    int e = (blockIdx.x * blockDim.x + threadIdx.x) >> 5;
    if (e >= E) return;
    int lane = threadIdx.x & 31;
    int t = ety[e], s = src[e], d = dst[e];
    float alpha = logit[e] / denom[d];
    float mv = vrel[((size_t)t * N + s) * HEADD + lane];
    atomicAdd(&agg[(size_t)d * HEADD + lane], alpha * mv);
}

// -------- out-proj + skip gate + ReLU: hn = relu(gate*(agg@Wa) + (1-g)*h) ----
__global__ void k_outproj(const float* __restrict__ agg, const float* __restrict__ Wa,
                          const float* __restrict__ h, const float* __restrict__ skip_l,
                          float* __restrict__ hnext, int N)
{
    int lane = threadIdx.x & 31;
    int tile = blockIdx.x * (blockDim.x >> 5) + (threadIdx.x >> 5);
    int row0 = tile * 16;
    if (row0 >= N) return;
    v8f c0 = wmma_tile_f32(agg, HEADD, row0, Wa, HIDD, 0, HEADD, lane);
    v8f c1 = wmma_tile_f32(agg, HEADD, row0, Wa, HIDD, 16, HEADD, lane);
    float gate = 1.f / (1.f + __expf(-skip_l[0]));
    float og = 1.f - gate;
    int n = lane & 15, hi = lane >> 4;
#pragma unroll
    for (int j = 0; j < 8; ++j) {
        int r = row0 + j + 8 * hi;
        float v0 = c0[j] * gate + h[(size_t)r * HIDD + n] * og;
        float v1 = c1[j] * gate + h[(size_t)r * HIDD + 16 + n] * og;
        hnext[(size_t)r * HIDD + n]      = fmaxf(v0, 0.f);
        hnext[(size_t)r * HIDD + 16 + n] = fmaxf(v1, 0.f);
    }
}

// -------- final classifier: out = h @ mlpW + mlpb  (32 -> 2) -----------------
__global__ void k_cls(const float* __restrict__ h, const float* __restrict__ W,
                      const float* __restrict__ b, float* __restrict__ out, int N, int C)
{
    int n = blockIdx.x * blockDim.x + threadIdx.x;
    if (n >= N) return;
    const float* hr = h + (size_t)n * HIDD;
    for (int c = 0; c < C; ++c) {
        float acc = b[c];
#pragma unroll
        for (int i = 0; i < HIDD; ++i) acc += hr[i] * W[i * C + c];
        out[(size_t)n * C + c] = acc;
    }
}

extern "C" void kernel_launch(void* const* d_in, const int* in_sizes, int n_in,
                              void* d_out, int out_size, void* d_ws, size_t ws_size,
                              hipStream_t stream)
{
    const float* x    = (const float*)d_in[0];
    const int*   src  = (const int*)d_in[1];
    const int*   dst  = (const int*)d_in[2];
    const int*   ety  = (const int*)d_in[3];
    const float* inW  = (const float*)d_in[4];
    const float* inb  = (const float*)d_in[5];
    const float* Wk   = (const float*)d_in[6];
    const float* Wq   = (const float*)d_in[7];
    const float* Wv   = (const float*)d_in[8];
    const float* Wa   = (const float*)d_in[9];
    const float* pri  = (const float*)d_in[10];
    const float* Ratt = (const float*)d_in[11];
    const float* Rmsg = (const float*)d_in[12];
    const float* skip = (const float*)d_in[13];
    const float* mlpW = (const float*)d_in[14];
    const float* mlpb = (const float*)d_in[15];

    const int N    = in_sizes[0] / INF_;
    const int E    = in_sizes[1];
    const int Lnum = in_sizes[13];
    const int T    = in_sizes[10] / Lnum;       // HEADS == 1
    const int C    = in_sizes[15];

    // ---- workspace layout (f32 unless noted) ----
    float* ws = (float*)d_ws;
    size_t nh = (size_t)N * HIDD;
    float*    h0    = ws;  ws += nh;
    float*    h1    = ws;  ws += nh;
    float*    q     = ws;  ws += nh;
    float*    krel  = ws;  ws += (size_t)T * nh;
    float*    vrel  = ws;  ws += (size_t)T * nh;
    float*    agg   = ws;  ws += nh;
    float*    logit = ws;  ws += (size_t)E;
    unsigned* amax  = (unsigned*)ws; ws += N;
    float*    denom = ws;  ws += N;
    float*    CW    = ws;  // 2*T*32*32 floats

    const int tiles    = (N + 15) / 16;
    const int projBlk  = 128;                       // 4 waves / block
    const int projGrid = (tiles + 3) / 4;
    const int edgeGrid = (E + 7) / 8;               // wave-per-edge, 8 waves/block
    const int initGrid = ((int)nh + 255) / 256;

    k_inproj<<<projGrid, projBlk, 0, stream>>>(x, inW, inb, h0, N);

    float* hc = h0;
    float* hn = h1;
    for (int l = 0; l < Lnum; ++l) {
        const float* Wk_l   = Wk   + (size_t)l * HIDD * HEADD;
        const float* Wq_l   = Wq   + (size_t)l * HIDD * HEADD;
        const float* Wv_l   = Wv   + (size_t)l * HIDD * HEADD;
        const float* Wa_l   = Wa   + (size_t)l * HEADD * HIDD;
        const float* Ratt_l = Ratt + (size_t)l * T * HIDD * HEADD;
        const float* Rmsg_l = Rmsg + (size_t)l * T * HIDD * HEADD;
        const float* pri_l  = pri  + (size_t)l * T;

        k_combine<<<2 * T, HIDD * HEADD, 0, stream>>>(Wk_l, Wv_l, Ratt_l, Rmsg_l, CW, T);
        k_init<<<initGrid, 256, 0, stream>>>(amax, denom, agg, N);
        k_proj7<<<dim3(projGrid, 1 + 2 * T), projBlk, 0, stream>>>(hc, Wq_l, CW, q, krel, vrel, N, T);
        k_edge_logit<<<edgeGrid, 256, 0, stream>>>(src, dst, ety, krel, q, pri_l, logit, amax, E, N);
        k_edge_exp<<<(E + 255) / 256, 256, 0, stream>>>(logit, dst, amax, denom, E);
        k_edge_agg<<<edgeGrid, 256, 0, stream>>>(src, dst, ety, vrel, logit, denom, agg, E, N);
        k_outproj<<<projGrid, projBlk, 0, stream>>>(agg, Wa_l, hc, skip + l, hn, N);

        float* tmp = hc; hc = hn; hn = tmp;
    }

    k_cls<<<(N + 255) / 256, 256, 0, stream>>>(hc, mlpW, mlpb, (float*)d_out, N, C);
}